// QVCClassifier_5987184410897
// MI455X (gfx1250) — compile-verified
//
#include <hip/hip_runtime.h>

#define NQ   16
#define NL   4
#define DIM  65536      // 2^16 amplitudes (real)
#define TPB  1024       // 32 waves (wave32)

typedef __attribute__((ext_vector_type(2))) float v2f;
typedef __attribute__((ext_vector_type(8))) float v8f;

// ======== GF(2) linear algebra of the CNOT ring (compile-time) ========
// Ring R (one layer of CNOTs, applied q=0..15 in order):
//   ctrl bit = 15-q, tgt bit = 15-((q+1)%16); if ctrl set, flip tgt.
// Each CNOT is linear over GF(2) => R is an invertible 16x16 bit-matrix.
// No data is moved for CNOTs: after l virtual rings, logical index v lives
// at physical slot R^{-l}(v). linv_col(l,b) = R^{-l} e_b.
constexpr unsigned linv_col(int l, int b) {
  unsigned v = 1u << b;
  for (int i = 0; i < l; ++i)
    for (int q = 15; q >= 0; --q) {            // R^{-1}: CNOTs in reverse order
      const int cb = 15 - q, tb = 15 - ((q + 1) & 15);
      if ((v >> cb) & 1) v ^= 1u << tb;
    }
  return v;
}
// Physical offset of logical fiber value f (logical bits shift..shift+3).
constexpr unsigned fiber_off(int l, int shift, int f) {
  unsigned r = 0;
  for (int t = 0; t < 4; ++t)
    if ((f >> t) & 1) r ^= linv_col(l, shift + t);
  return r;
}
// Row 15 of R^4: sign of the measured qubit expressed in physical coords.
constexpr unsigned sign_mask4() {
  unsigned mask = 0;
  for (int b = 0; b < 16; ++b) {
    unsigned v = 1u << b;
    for (int k = 0; k < 4; ++k)
      for (int q = 0; q < 16; ++q) {           // forward ring, 4 times
        const int cb = 15 - q, tb = 15 - ((q + 1) & 15);
        if ((v >> cb) & 1) v ^= 1u << tb;
      }
    if ((v >> 15) & 1) mask |= 1u << b;
  }
  return mask;
}

struct FiberTab { unsigned v[16]; };
struct BaseTab  { unsigned v[12]; };
constexpr FiberTab make_foff(int l, int shift) {
  FiberTab r{};
  for (int f = 0; f < 16; ++f) r.v[f] = fiber_off(l, shift, f);
  return r;
}
// n-bit i (12 non-fiber bits of the fiber id) maps to logical bit (i<shift ? i : i+4)
constexpr BaseTab make_cn(int l, int shift) {
  BaseTab r{};
  for (int i = 0; i < 12; ++i) r.v[i] = linv_col(l, (i < shift) ? i : i + 4);
  return r;
}

// RY matrix element: [[c,-s],[s,c]][ob][ib]
__device__ __forceinline__ float ry_elem(float c, float s, int ob, int ib) {
  float v = (ob == ib) ? c : s;
  return ((ob == 0) & (ib == 1)) ? -v : v;
}
// Element (m,k) of U = RY(q0) (x) RY(q1) (x) RY(q2) (x) RY(q3)
__device__ __forceinline__ float u4_elem(int m, int k,
    float c0, float s0, float c1, float s1,
    float c2, float s2, float c3, float s3) {
  float r = ry_elem(c0, s0, (m >> 3) & 1, (k >> 3) & 1);
  r *= ry_elem(c1, s1, (m >> 2) & 1, (k >> 2) & 1);
  r *= ry_elem(c2, s2, (m >> 1) & 1, (k >> 1) & 1);
  r *= ry_elem(c3, s3,  m       & 1,  k       & 1);
  return r;
}

// One fused 16x16 Kronecker RY gate on qubits 4G..4G+3 of layer L,
// with L prior CNOT rings folded into compile-time address masks.
template<int L, int G>
__device__ __forceinline__ void gate_pass(float* __restrict__ S,
                                          const float* __restrict__ csT,
                                          int col, int hi16, int wave) {
  constexpr int shift = 12 - 4 * G;            // logical fiber bits shift..shift+3
  constexpr FiberTab FO = make_foff(L, shift); // forced compile-time
  constexpr BaseTab  CN = make_cn(L, shift);   // forced compile-time

  const int qb = 2 * (L * NQ + 4 * G);
  const float c0 = csT[qb + 0], s0 = csT[qb + 1];
  const float c1 = csT[qb + 2], s1 = csT[qb + 3];
  const float c2 = csT[qb + 4], s2 = csT[qb + 5];
  const float c3 = csT[qb + 6], s3 = csT[qb + 7];

  // A (16x4 f32): lanes 0-15 -> M=lane, K=k0+{0,1}; lanes 16-31 -> K=k0+{2,3}
  const int kk = hi16 << 1;                    // 0 or 2
  v2f A0, A1, A2, A3;
  A0.x = u4_elem(col,  0 + kk, c0,s0,c1,s1,c2,s2,c3,s3);
  A0.y = u4_elem(col,  1 + kk, c0,s0,c1,s1,c2,s2,c3,s3);
  A1.x = u4_elem(col,  4 + kk, c0,s0,c1,s1,c2,s2,c3,s3);
  A1.y = u4_elem(col,  5 + kk, c0,s0,c1,s1,c2,s2,c3,s3);
  A2.x = u4_elem(col,  8 + kk, c0,s0,c1,s1,c2,s2,c3,s3);
  A2.y = u4_elem(col,  9 + kk, c0,s0,c1,s1,c2,s2,c3,s3);
  A3.x = u4_elem(col, 12 + kk, c0,s0,c1,s1,c2,s2,c3,s3);
  A3.y = u4_elem(col, 13 + kk, c0,s0,c1,s1,c2,s2,c3,s3);

  // Per-lane physical fiber offsets: selects between literal constants.
  unsigned FL[8];
#pragma unroll
  for (int kc = 0; kc < 4; ++kc)
#pragma unroll
    for (int e = 0; e < 2; ++e)
      FL[kc * 2 + e] = hi16 ? FO.v[kc * 4 + 2 + e] : FO.v[kc * 4 + e];
  unsigned FS[8];
#pragma unroll
  for (int v = 0; v < 8; ++v)
    FS[v] = hi16 ? FO.v[v + 8] : FO.v[v];

  // Fixed part of the physical tile base: n = {wave[4:0], tt[2:0], col[3:0]}
  unsigned baseF = 0;
#pragma unroll
  for (int i = 0; i < 4; ++i)
    baseF ^= ((col >> i) & 1) ? CN.v[i] : 0u;
#pragma unroll
  for (int w = 0; w < 5; ++w)
    baseF ^= ((wave >> w) & 1) ? CN.v[7 + w] : 0u;

#pragma unroll
  for (int tt = 0; tt < 8; ++tt) {
    const unsigned base = baseF ^ ((tt & 1) ? CN.v[4] : 0u)
                                ^ ((tt & 2) ? CN.v[5] : 0u)
                                ^ ((tt & 4) ? CN.v[6] : 0u);  // folds to 1 literal
    // B (4x16 f32): lanes 0-15 supply K=k0+{0,1}, lanes 16-31 K=k0+{2,3}
    v2f B0, B1, B2, B3;
    B0.x = S[base ^ FL[0]];  B0.y = S[base ^ FL[1]];
    B1.x = S[base ^ FL[2]];  B1.y = S[base ^ FL[3]];
    B2.x = S[base ^ FL[4]];  B2.y = S[base ^ FL[5]];
    B3.x = S[base ^ FL[6]];  B3.y = S[base ^ FL[7]];

    v8f acc = {};
    acc = __builtin_amdgcn_wmma_f32_16x16x4_f32(false, A0, false, B0, (short)0, acc, false, false);
    acc = __builtin_amdgcn_wmma_f32_16x16x4_f32(false, A1, false, B1, (short)0, acc, false, false);
    acc = __builtin_amdgcn_wmma_f32_16x16x4_f32(false, A2, false, B2, (short)0, acc, false, false);
    acc = __builtin_amdgcn_wmma_f32_16x16x4_f32(false, A3, false, B3, (short)0, acc, false, false);

    // D (16x16 f32): lanes 0-15 rows M=v, lanes 16-31 rows M=v+8
#pragma unroll
    for (int v = 0; v < 8; ++v)
      S[base ^ FS[v]] = acc[v];
  }
}

__global__ __launch_bounds__(TPB, 1)
void qvc16_kernel(const float* __restrict__ x, const float* __restrict__ theta,
                  const float* __restrict__ head_w, const float* __restrict__ head_b,
                  float* __restrict__ out) {
  extern __shared__ float lds[];
  float* S   = lds;          // [DIM]   state vector (256 KB, LDS-resident)
  float* csX = lds + DIM;    // [32]    (cos,sin) of x/2 per qubit
  float* csT = csX + 32;     // [128]   (cos,sin) of theta/2 per angle
  float* red = csT + 128;    // [1]     reduction cell

  const int b    = blockIdx.x;
  const int tid  = threadIdx.x;
  const int lane = tid & 31;
  const int wave = tid >> 5;
  const int col  = lane & 15;
  const int hi16 = lane >> 4;          // 0 or 1

  if (tid == 0) {
    red[0] = 0.0f;
    __builtin_prefetch(x + b * NQ, 0, 0);   // global_prefetch_b8
  }
  if (tid < NQ) {
    float a = 0.5f * x[b * NQ + tid];
    float s, c; sincosf(a, &s, &c);
    csX[2 * tid] = c; csX[2 * tid + 1] = s;
  } else if (tid < NQ + NL * NQ) {
    int k = tid - NQ;
    float a = 0.5f * theta[k];
    float s, c; sincosf(a, &s, &c);
    csT[2 * k] = c; csT[2 * k + 1] = s;
  }
  __syncthreads();

  // ---- Product-state init: state after RY(x_q) on |0...0>, qubit q <-> bit 15-q
  for (int i = tid; i < DIM; i += TPB) {
    float amp = 1.0f;
#pragma unroll
    for (int q = 0; q < NQ; ++q)
      amp *= csX[2 * q + ((i >> (15 - q)) & 1)];
    S[i] = amp;
  }
  __syncthreads();

  // ---- 16 barriered WMMA gate passes; CNOT rings are purely virtual.
  gate_pass<0,0>(S, csT, col, hi16, wave); __syncthreads();
  gate_pass<0,1>(S, csT, col, hi16, wave); __syncthreads();
  gate_pass<0,2>(S, csT, col, hi16, wave); __syncthreads();
  gate_pass<0,3>(S, csT, col, hi16, wave); __syncthreads();
  gate_pass<1,0>(S, csT, col, hi16, wave); __syncthreads();
  gate_pass<1,1>(S, csT, col, hi16, wave); __syncthreads();
  gate_pass<1,2>(S, csT, col, hi16, wave); __syncthreads();
  gate_pass<1,3>(S, csT, col, hi16, wave); __syncthreads();
  gate_pass<2,0>(S, csT, col, hi16, wave); __syncthreads();
  gate_pass<2,1>(S, csT, col, hi16, wave); __syncthreads();
  gate_pass<2,2>(S, csT, col, hi16, wave); __syncthreads();
  gate_pass<2,3>(S, csT, col, hi16, wave); __syncthreads();
  gate_pass<3,0>(S, csT, col, hi16, wave); __syncthreads();
  gate_pass<3,1>(S, csT, col, hi16, wave); __syncthreads();
  gate_pass<3,2>(S, csT, col, hi16, wave); __syncthreads();
  gate_pass<3,3>(S, csT, col, hi16, wave); __syncthreads();

  // ---- z0 = sum(+-amp^2); sign = bit15 of logical index = parity(p & row15(R^4))
  constexpr unsigned SM = sign_mask4();
  float part = 0.0f;
  for (int i = tid; i < DIM; i += TPB) {
    float v = S[i];
    float sq = v * v;
    part += (__popc((unsigned)i & SM) & 1) ? -sq : sq;
  }
#pragma unroll
  for (int off = 16; off > 0; off >>= 1)
    part += __shfl_down(part, off, 32);
  if (lane == 0) atomicAdd(red, part);         // ds_add_f32
  __syncthreads();
  if (tid == 0) out[b] = head_w[0] * red[0] + head_b[0];
}

extern "C" void kernel_launch(void* const* d_in, const int* in_sizes, int n_in,
                              void* d_out, int out_size, void* d_ws, size_t ws_size,
                              hipStream_t stream) {
  const float* x      = (const float*)d_in[0];
  const float* theta  = (const float*)d_in[1];
  const float* head_w = (const float*)d_in[2];
  const float* head_b = (const float*)d_in[3];
  float* out = (float*)d_out;
  const int B = in_sizes[0] / NQ;              // 1024
  const size_t shmem = (size_t)(DIM + 32 + 128 + 4) * sizeof(float); // ~256.7 KB < 320 KB
  qvc16_kernel<<<B, TPB, shmem, stream>>>(x, theta, head_w, head_b, out);
}